// NeuralOscillator_2241972929164
// MI455X (gfx1250) — compile-verified
//
#include <hip/hip_runtime.h>
#include <hip/hip_bf16.h>

typedef __attribute__((ext_vector_type(2))) float v2f;
typedef __attribute__((ext_vector_type(4))) float v4f;
typedef __attribute__((ext_vector_type(8))) float v8f;

#define WSTRIDE 68              // padded LDS row stride (floats) -> conflict-free frag loads
#define WLAYER  (64 * WSTRIDE)  // one 64x64 weight matrix, padded
#define H_OSC   0.1f
#define H_WAVE  0.1f

// ---------------- fp32 WMMA wrapper: D = A(16x4) x B(4x16) + C -----------------
__device__ __forceinline__ v8f wmma4(v2f a, v2f b, v8f c) {
  // 8 args: (neg_a, A, neg_b, B, c_mod, C, reuse_a, reuse_b)
  return __builtin_amdgcn_wmma_f32_16x16x4_f32(false, a, false, b, (short)0, c,
                                               false, false);
}

// ---------------- small utility kernels -----------------
__global__ void zero_f4(float4* p, size_t n4) {
  size_t i = (size_t)blockIdx.x * blockDim.x + threadIdx.x;
  size_t stride = (size_t)gridDim.x * blockDim.x;
  float4 z = {0.f, 0.f, 0.f, 0.f};
  for (; i < n4; i += stride) p[i] = z;
}

__global__ void zero_i32(int* p, int n) {
  int i = blockIdx.x * blockDim.x + threadIdx.x;
  if (i < n) p[i] = 0;
}

__global__ void count_deg(const int* __restrict__ col, int* cnt, int E) {
  int e = blockIdx.x * blockDim.x + threadIdx.x;
  if (e < E) atomicAdd(&cnt[col[e]], 1);
}

#define SCAN_T 1024
__global__ void build_rowptr(int* cnt, int* row_ptr, float* degf, int n) {
  __shared__ int buf[SCAN_T];
  __shared__ int carry;
  if (threadIdx.x == 0) carry = 0;
  __syncthreads();
  int nChunks = (n + SCAN_T - 1) / SCAN_T;
  for (int ch = 0; ch < nChunks; ++ch) {
    int idx = ch * SCAN_T + threadIdx.x;
    int c = 0;
    if (idx < n) {
      c = cnt[idx];
      degf[idx] = (float)c;
      cnt[idx] = 0;  // becomes the fill cursor
    }
    buf[threadIdx.x] = c;
    __syncthreads();
    for (int off = 1; off < SCAN_T; off <<= 1) {
      int v = 0;
      if ((int)threadIdx.x >= off) v = buf[threadIdx.x - off];
      __syncthreads();
      if ((int)threadIdx.x >= off) buf[threadIdx.x] += v;
      __syncthreads();
    }
    if (idx < n) row_ptr[idx] = carry + buf[threadIdx.x] - c;  // exclusive
    __syncthreads();
    if (threadIdx.x == 0) carry += buf[SCAN_T - 1];
    __syncthreads();
  }
  if (threadIdx.x == 0) row_ptr[n] = carry;
}

__global__ void fill_nbr(const int* __restrict__ row, const int* __restrict__ col,
                         const int* __restrict__ row_ptr, int* cursor, int* nbr, int E) {
  int e = blockIdx.x * blockDim.x + threadIdx.x;
  if (e < E) {
    int c = col[e];
    int slot = atomicAdd(&cursor[c], 1);
    nbr[row_ptr[c] + slot] = row[e];
  }
}

// canonical per-bucket order => deterministic float summation across replays
__global__ void sort_nbrs(int* nbr, const int* __restrict__ row_ptr, int n) {
  int i = blockIdx.x * blockDim.x + threadIdx.x;
  if (i >= n) return;
  int s = row_ptr[i], e = row_ptr[i + 1];
  for (int a = s + 1; a < e; ++a) {
    int v = nbr[a];
    int b = a - 1;
    while (b >= s && nbr[b] > v) { nbr[b + 1] = nbr[b]; --b; }
    nbr[b + 1] = v;
  }
}

__global__ void embed_kernel(const int* __restrict__ x, const float* __restrict__ key_emb,
                             const float* __restrict__ val_emb, float* __restrict__ Xe, int n) {
  long long t = (long long)blockIdx.x * blockDim.x + threadIdx.x;  // n*16 threads
  if (t >= (long long)n * 16) return;
  int node = (int)(t >> 4);
  int dp = (int)(t & 15) * 4;
  int k = x[node * 2 + 0], v = x[node * 2 + 1];
  v4f a = *(const v4f*)&key_emb[k * 64 + dp];
  v4f b = *(const v4f*)&val_emb[v * 64 + dp];
  *(v4f*)&Xe[(size_t)node * 64 + dp] = a + b;
}

// copy a 16x64 row-major tile from global to padded LDS (stride WSTRIDE)
// 32 lanes move 2 rows (512B fully coalesced) per iteration via b128
__device__ __forceinline__ void copy_tile_to_lds(const float* __restrict__ g, float* scr,
                                                 int base, int lane) {
  const int l16 = lane & 15;
  const int rh = lane >> 4;
#pragma unroll
  for (int m2 = 0; m2 < 8; ++m2) {
    const int row = m2 * 2 + rh;
    v4f t = *(const v4f*)&g[(size_t)(base + row) * 64 + 4 * l16];
    *(v4f*)&scr[row * WSTRIDE + 4 * l16] = t;
  }
}

// One oscillator layer for a 16-node tile.
//   acc = Asrc(16x64) @ W^T (W rows staged in LDS, padded)
//   u' = u + h*relu(a*z + acc + b); z' = z + h*u'   (Z/U updated in place)
__device__ __forceinline__ void osc_layer(int lane, int base,
                                          const float* __restrict__ Asrc,   // LDS tile
                                          float* __restrict__ scrOut,       // LDS out (or null)
                                          float* Zj, float* Uj,             // global [n][64]
                                          const float* __restrict__ wBwL,   // LDS 64xWSTRIDE
                                          const float* __restrict__ aRow,
                                          const float* __restrict__ bRow, bool writeScr) {
  const int lane16 = lane & 15;
  const int laneHi = lane >> 4;
  v8f acc0 = {0, 0, 0, 0, 0, 0, 0, 0};
  v8f acc1 = acc0, acc2 = acc0, acc3 = acc0;
  const float* arow = Asrc + lane16 * WSTRIDE + laneHi * 2;
  const float* b0r = wBwL + (0 * 16 + lane16) * WSTRIDE + laneHi * 2;
  const float* b1r = b0r + 16 * WSTRIDE;
  const float* b2r = b1r + 16 * WSTRIDE;
  const float* b3r = b2r + 16 * WSTRIDE;
#pragma unroll
  for (int kt = 0; kt < 16; ++kt) {
    const int kk = kt * 4;
    v2f a = *(const v2f*)(arow + kk);
    v2f b0 = *(const v2f*)(b0r + kk);
    v2f b1 = *(const v2f*)(b1r + kk);
    v2f b2 = *(const v2f*)(b2r + kk);
    v2f b3 = *(const v2f*)(b3r + kk);
    acc0 = wmma4(a, b0, acc0);
    acc1 = wmma4(a, b1, acc1);
    acc2 = wmma4(a, b2, acc2);
    acc3 = wmma4(a, b3, acc3);
  }
  v8f accs[4] = {acc0, acc1, acc2, acc3};
#pragma unroll
  for (int nt = 0; nt < 4; ++nt) {
    const int col = nt * 16 + lane16;
    const float av = aRow[col];
    const float bv = bRow[col];
#pragma unroll
    for (int vr = 0; vr < 8; ++vr) {
      const int m = vr + laneHi * 8;
      const size_t gi = (size_t)(base + m) * 64 + col;
      float z = Zj[gi];
      float u = Uj[gi];
      float pre = fmaf(av, z, accs[nt][vr] + bv);
      float r = fmaxf(pre, 0.0f);
      float un = fmaf(H_OSC, r, u);
      float zn = fmaf(H_OSC, un, z);
      Uj[gi] = un;
      Zj[gi] = zn;
      if (writeScr) scrOut[m * WSTRIDE + col] = zn;
    }
  }
}

// ---------------- initial encoder: X0 = relu(Xe @ W^T + b) -----------------
__global__ void init_gemm(const float* __restrict__ Xe, float* __restrict__ X0,
                          const float* __restrict__ W, const float* __restrict__ bias, int n) {
  extern __shared__ float smem[];
  float* wW = smem;                    // 64 * WSTRIDE
  float* wBias = wW + WLAYER;          // 64
  float* scr = wBias + 64;             // per-wave 16 * WSTRIDE
  const int wave = threadIdx.x >> 5, lane = threadIdx.x & 31;

  for (int i = threadIdx.x; i < 64 * 16; i += blockDim.x) {
    int r = i >> 4, c4 = i & 15;
    float4 v = *(const float4*)&W[r * 64 + c4 * 4];
    *(float4*)&wW[r * WSTRIDE + c4 * 4] = v;
  }
  for (int i = threadIdx.x; i < 64; i += blockDim.x) wBias[i] = bias[i];
  __syncthreads();

  int tile = blockIdx.x * (blockDim.x >> 5) + wave;
  int base = tile * 16;
  if (base >= n) return;
  float* myscr = scr + wave * 16 * WSTRIDE;
  copy_tile_to_lds(Xe, myscr, base, lane);

  const int lane16 = lane & 15, laneHi = lane >> 4;
  v8f acc0 = {0, 0, 0, 0, 0, 0, 0, 0};
  v8f acc1 = acc0, acc2 = acc0, acc3 = acc0;
  const float* arow = myscr + lane16 * WSTRIDE + laneHi * 2;
  const float* b0r = wW + (0 * 16 + lane16) * WSTRIDE + laneHi * 2;
  const float* b1r = b0r + 16 * WSTRIDE;
  const float* b2r = b1r + 16 * WSTRIDE;
  const float* b3r = b2r + 16 * WSTRIDE;
#pragma unroll
  for (int kt = 0; kt < 16; ++kt) {
    const int kk = kt * 4;
    v2f a = *(const v2f*)(arow + kk);
    acc0 = wmma4(a, *(const v2f*)(b0r + kk), acc0);
    acc1 = wmma4(a, *(const v2f*)(b1r + kk), acc1);
    acc2 = wmma4(a, *(const v2f*)(b2r + kk), acc2);
    acc3 = wmma4(a, *(const v2f*)(b3r + kk), acc3);
  }
  v8f accs[4] = {acc0, acc1, acc2, acc3};
#pragma unroll
  for (int nt = 0; nt < 4; ++nt) {
    const int col = nt * 16 + lane16;
    const float bv = wBias[col];
#pragma unroll
    for (int vr = 0; vr < 8; ++vr) {
      const int m = vr + laneHi * 8;
      X0[(size_t)(base + m) * 64 + col] = fmaxf(accs[nt][vr] + bv, 0.0f);
    }
  }
}

// ---------------- fused time-step kernel -----------------
// Per 16-node wave tile: Laplacian leapfrog (X double-buffered), then the 4
// oscillator layers in reference order (0, 3, 2, 1) so "old" Z2/Z1 and "new"
// X/Z0 semantics match exactly. All cross-node reads are of OLD X only.
__global__ void step_kernel(const float* __restrict__ Xcur, float* __restrict__ Xnext,
                            float* __restrict__ Y, float* Z, float* U,
                            const float* __restrict__ degf, const int* __restrict__ row_ptr,
                            const int* __restrict__ nbr, const float* __restrict__ osc_a,
                            const float* __restrict__ osc_Bw, const float* __restrict__ osc_Bb,
                            int n) {
  extern __shared__ float smem[];
  float* wBw = smem;                    // 4 * WLAYER
  float* wAv = wBw + 4 * WLAYER;        // 4*64
  float* wBb = wAv + 256;               // 4*64
  float* scr = wBb + 256;               // per-wave 2 * 16 * WSTRIDE
  const int wave = threadIdx.x >> 5, lane = threadIdx.x & 31;

  // stage oscillator weights into LDS (padded rows)
  for (int i = threadIdx.x; i < 4 * 64 * 16; i += blockDim.x) {
    int r = i >> 4, c4 = i & 15;
    float4 v = *(const float4*)&osc_Bw[r * 64 + c4 * 4];
    *(float4*)&wBw[r * WSTRIDE + c4 * 4] = v;
  }
  for (int i = threadIdx.x; i < 256; i += blockDim.x) {
    wAv[i] = osc_a[i];
    wBb[i] = osc_Bb[i];
  }
  __syncthreads();

  int tile = blockIdx.x * (blockDim.x >> 5) + wave;
  int base = tile * 16;
  if (base >= n) return;

  float* scrA = scr + wave * 2 * 16 * WSTRIDE;  // xnew, then recycled for old Z tiles
  float* scrB = scrA + 16 * WSTRIDE;            // new z0 (A-source of layer 1)
  const size_t n64 = (size_t)n * 64;

  // ---- Phase A: Laplacian + leapfrog ----
  // Half-waves process even/odd neighbors with b128 row loads (2 rows = 512B
  // per trip); partials combined with a cross-half shuffle (deterministic).
  const int half = lane >> 4;
  const int l16 = lane & 15;
  const int d4 = 4 * l16;  // each lane covers dims d4..d4+3
  for (int m = 0; m < 16; ++m) {
    const int node = base + m;
    const int s = row_ptr[node], e = row_ptr[node + 1];
    v4f agg = {0.f, 0.f, 0.f, 0.f};
    for (int p = s + half; p < e; p += 2) {
      if (p + 4 < e) __builtin_prefetch(&Xcur[(size_t)nbr[p + 4] * 64 + d4], 0, 0);
      agg += *(const v4f*)&Xcur[(size_t)nbr[p] * 64 + d4];
    }
    v4f tot;
    tot.x = agg.x + __shfl_xor(agg.x, 16, 32);
    tot.y = agg.y + __shfl_xor(agg.y, 16, 32);
    tot.z = agg.z + __shfl_xor(agg.z, 16, 32);
    tot.w = agg.w + __shfl_xor(agg.w, 16, 32);
    if (half == 0) {  // lanes 0-15 hold the full row sum; do the leapfrog
      const size_t gi = (size_t)node * 64 + d4;
      v4f xo = *(const v4f*)&Xcur[gi];
      v4f yo = *(const v4f*)&Y[gi];
      const float dg = degf[node];
      v4f vel = xo * dg - tot;
      yo = yo - vel * H_WAVE;
      v4f xn = xo + yo * H_WAVE;
      *(v4f*)&Y[gi] = yo;
      *(v4f*)&Xnext[gi] = xn;
      *(v4f*)&scrA[m * WSTRIDE + d4] = xn;
    }
  }

  // ---- Phase B: oscillator stack (order 0, 3, 2, 1) ----
  // layer 0: z_prev = new X (scrA); produce new z0 into scrB
  osc_layer(lane, base, scrA, scrB, Z + 0 * n64, U + 0 * n64, wBw + 0 * WLAYER, wAv + 0,
            wBb + 0, true);
  // layer 3: z_prev = old Z2
  copy_tile_to_lds(Z + 2 * n64, scrA, base, lane);
  osc_layer(lane, base, scrA, nullptr, Z + 3 * n64, U + 3 * n64, wBw + 3 * WLAYER, wAv + 192,
            wBb + 192, false);
  // layer 2: z_prev = old Z1
  copy_tile_to_lds(Z + 1 * n64, scrA, base, lane);
  osc_layer(lane, base, scrA, nullptr, Z + 2 * n64, U + 2 * n64, wBw + 2 * WLAYER, wAv + 128,
            wBb + 128, false);
  // layer 1: z_prev = NEW z0 (scrB)
  osc_layer(lane, base, scrB, nullptr, Z + 1 * n64, U + 1 * n64, wBw + 1 * WLAYER, wAv + 64,
            wBb + 64, false);
}

// ---------------- readout: out[r][o] = dot(Z3[root[r]], final_w[o]) -----------------
__global__ void final_kernel(const float* __restrict__ Z3, const int* __restrict__ root,
                             const float* __restrict__ Wf, float* __restrict__ out, int R,
                             int ncols) {
  const int wave = threadIdx.x >> 5, lane = threadIdx.x & 31;
  const int r = blockIdx.x * (blockDim.x >> 5) + wave;
  if (r >= R) return;
  const int node = root[r];
  const float z0 = Z3[(size_t)node * 64 + lane];
  const float z1 = Z3[(size_t)node * 64 + 32 + lane];
  for (int o = 0; o < ncols; ++o) {
    float s = fmaf(z0, Wf[o * 64 + lane], z1 * Wf[o * 64 + 32 + lane]);
    for (int off = 16; off > 0; off >>= 1) s += __shfl_xor(s, off, 32);
    if (lane == 0) out[r * ncols + o] = s;
  }
}

// ---------------- launcher -----------------
extern "C" void kernel_launch(void* const* d_in, const int* in_sizes, int n_in, void* d_out,
                              int out_size, void* d_ws, size_t ws_size, hipStream_t stream) {
  (void)n_in; (void)out_size; (void)ws_size;
  const int n = in_sizes[0] / 2;
  const int E = in_sizes[1] / 2;
  const int R = in_sizes[2];
  const int ncols = in_sizes[10] / 64;

  const int* x = (const int*)d_in[0];
  const int* ei = (const int*)d_in[1];
  const int* root = (const int*)d_in[2];
  const float* key_emb = (const float*)d_in[3];
  const float* val_emb = (const float*)d_in[4];
  const float* lin_w = (const float*)d_in[5];
  const float* lin_b = (const float*)d_in[6];
  const float* osc_a = (const float*)d_in[7];
  const float* osc_Bw = (const float*)d_in[8];
  const float* osc_Bb = (const float*)d_in[9];
  const float* fin_w = (const float*)d_in[10];
  float* out = (float*)d_out;

  const size_t n64 = (size_t)n * 64;
  float* X0 = (float*)d_ws;
  float* X1 = X0 + n64;
  float* Y = X1 + n64;
  float* Z = Y + n64;
  float* U = Z + 4 * n64;
  float* degf = U + 4 * n64;
  int* cnt = (int*)(degf + n);
  int* row_ptr = cnt + n;
  int* nbr = row_ptr + (n + 1);

  const int T = 256;
  // zero Y, Z, U (contiguous: 9*n64 floats)
  zero_f4<<<4096, T, 0, stream>>>((float4*)Y, (9 * n64) / 4);
  zero_i32<<<(n + T - 1) / T, T, 0, stream>>>(cnt, n);
  // deterministic CSR build
  count_deg<<<(E + T - 1) / T, T, 0, stream>>>(ei + E, cnt, E);
  build_rowptr<<<1, SCAN_T, 0, stream>>>(cnt, row_ptr, degf, n);
  fill_nbr<<<(E + T - 1) / T, T, 0, stream>>>(ei, ei + E, row_ptr, cnt, nbr, E);
  sort_nbrs<<<(n + T - 1) / T, T, 0, stream>>>(nbr, row_ptr, n);
  // encoder
  embed_kernel<<<(int)(((long long)n * 16 + T - 1) / T), T, 0, stream>>>(x, key_emb, val_emb,
                                                                         X1, n);
  const int tiles = (n + 15) / 16;
  const int gblocks = (tiles + 7) / 8;
  const int INIT_SMEM = (WLAYER + 64 + 8 * 16 * WSTRIDE) * 4;
  const int STEP_SMEM = (4 * WLAYER + 512 + 8 * 2 * 16 * WSTRIDE) * 4;
  hipFuncSetAttribute((const void*)init_gemm, hipFuncAttributeMaxDynamicSharedMemorySize,
                      INIT_SMEM);
  hipFuncSetAttribute((const void*)step_kernel, hipFuncAttributeMaxDynamicSharedMemorySize,
                      STEP_SMEM);
  init_gemm<<<gblocks, T, INIT_SMEM, stream>>>(X1, X0, lin_w, lin_b, n);

  float* Xc = X0;
  float* Xn = X1;
  for (int s = 0; s < 8; ++s) {
    step_kernel<<<gblocks, T, STEP_SMEM, stream>>>(Xc, Xn, Y, Z, U, degf, row_ptr, nbr, osc_a,
                                                   osc_Bw, osc_Bb, n);
    float* t = Xc; Xc = Xn; Xn = t;
  }
  final_kernel<<<(R + 7) / 8, T, 0, stream>>>(Z + 3 * n64, root, fin_w, out, R, ncols);
}